// ResnetBlockFC_32968168964592
// MI455X (gfx1250) — compile-verified
//
#include <hip/hip_runtime.h>

typedef float v2f __attribute__((ext_vector_type(2)));
typedef float v8f __attribute__((ext_vector_type(8)));

#define NN 40962
#define NE 245760
#define KG 10
#define GEPS 1e-15f

// ---------------------------------------------------------------- utilities
__global__ void zero_f32(float* __restrict__ p, size_t n) {
  size_t i = (size_t)blockIdx.x * blockDim.x + threadIdx.x;
  size_t st = (size_t)gridDim.x * blockDim.x;
  for (; i < n; i += st) p[i] = 0.0f;
}

// ------------------------------------------- gaussian edge weights + degree
__device__ __forceinline__ float gweight(float p0, float p1,
                                         const float* __restrict__ mu,
                                         const float* __restrict__ sg, int k) {
  float d0 = p0 - mu[2 * k + 0];
  float d1 = p1 - mu[2 * k + 1];
  float s0 = sg[2 * k + 0];
  float s1 = sg[2 * k + 1];
  return expf(-0.5f * (d0 * d0 / (GEPS + s0 * s0) + d1 * d1 / (GEPS + s1 * s1)));
}

__global__ void edge_weights_deg(const float* __restrict__ pseudo,
                                 const int* __restrict__ dst,
                                 const float* __restrict__ mu1, const float* __restrict__ sg1,
                                 const float* __restrict__ mu2, const float* __restrict__ sg2,
                                 const float* __restrict__ mus, const float* __restrict__ sgs,
                                 float* __restrict__ w1, float* __restrict__ w2,
                                 float* __restrict__ ws, float* __restrict__ deg) {
  int e = blockIdx.x * blockDim.x + threadIdx.x;
  if (e >= NE) return;
  float p0 = pseudo[2 * e + 0];
  float p1 = pseudo[2 * e + 1];
#pragma unroll
  for (int k = 0; k < KG; ++k) {
    w1[(size_t)e * KG + k] = gweight(p0, p1, mu1, sg1, k);
    w2[(size_t)e * KG + k] = gweight(p0, p1, mu2, sg2, k);
    ws[(size_t)e * KG + k] = gweight(p0, p1, mus, sgs, k);
  }
  atomicAdd(&deg[dst[e]], 1.0f);
}

// ------------------------------------------------ fp32 WMMA GEMM, fixed K=64
// C[M,NC] = A[M,64] @ B[64,NC], NC compile-time (immediate B offsets).
// One wave32 computes a 32x16 slab of C (two 16x16 tiles sharing B fragments)
// via 2x16 V_WMMA_F32_16X16X4_F32 as two independent accumulator chains.
// A 16x4 f32 layout: lane = m + 16*(k/2), vgpr = k%2.  B 4x16 mirrors it.
// C/D: vgpr v holds row base + 8*(lane/16) + v, col = tn*16 + lane%16.
// Out-of-range A rows are clamped (branchless); they only pollute C rows that
// are never stored.
template <int NC>
__global__ void gemm_k64_wmma_f32(const float* __restrict__ A,
                                  const float* __restrict__ B,
                                  float* __restrict__ C, int M) {
  constexpr int tilesN = NC / 16;
  const int lane = threadIdx.x & 31;
  const int wave = threadIdx.x >> 5;
  const int pairsM = (M + 31) >> 5;
  const int tile = blockIdx.x * (blockDim.x >> 5) + wave;
  if (tile >= pairsM * tilesN) return;  // wave-uniform exit (EXEC stays full)
  const int tp = tile / tilesN;
  const int tn = tile - tp * tilesN;

  const int mloc = lane & 15;
  const int khalf = (lane >> 4) << 1;  // 0 or 2
  const int row0 = tp * 32 + mloc;
  const int row1 = row0 + 16;
  const int col = tn * 16 + mloc;

  const int r0c = row0 < M ? row0 : M - 1;  // branchless clamp for loads
  const int r1c = row1 < M ? row1 : M - 1;

  const float* A0 = A + (size_t)r0c * 64 + khalf;
  const float* A1 = A + (size_t)r1c * 64 + khalf;
  const float* Bp = B + (size_t)khalf * NC + col;

  v8f acc0 = {0.f, 0.f, 0.f, 0.f, 0.f, 0.f, 0.f, 0.f};
  v8f acc1 = {0.f, 0.f, 0.f, 0.f, 0.f, 0.f, 0.f, 0.f};
#pragma unroll
  for (int k0 = 0; k0 < 64; k0 += 4) {
    v2f b;
    b[0] = Bp[k0 * NC];
    b[1] = Bp[(k0 + 1) * NC];
    v2f a0;
    a0[0] = A0[k0];
    a0[1] = A0[k0 + 1];
    v2f a1;
    a1[0] = A1[k0];
    a1[1] = A1[k0 + 1];
    acc0 = __builtin_amdgcn_wmma_f32_16x16x4_f32(false, a0, false, b,
                                                 (short)0, acc0, false, false);
    acc1 = __builtin_amdgcn_wmma_f32_16x16x4_f32(false, a1, false, b,
                                                 (short)0, acc1, false, false);
  }

  const int rbase0 = tp * 32 + ((lane >> 4) << 3);
#pragma unroll
  for (int v = 0; v < 8; ++v) {
    int r = rbase0 + v;
    if (r < M) C[(size_t)r * NC + col] = acc0[v];
  }
#pragma unroll
  for (int v = 0; v < 8; ++v) {
    int r = rbase0 + 16 + v;
    if (r < M) C[(size_t)r * NC + col] = acc1[v];
  }
}

// ---------------------------- edge stage: gather Z[src], K-weighted sum, scatter
// blockDim.x == Cout (compile-time -> immediate gather offsets).  Grid-stride
// over edges with one-iteration lookahead prefetch of the next random Z row.
template <int Cout>
__global__ void edge_aggregate(const float* __restrict__ Z, const float* __restrict__ W,
                               const int* __restrict__ src, const int* __restrict__ dst,
                               float* __restrict__ agg) {
  const int c = threadIdx.x;
  constexpr int KC = KG * Cout;
  for (int e = blockIdx.x; e < NE; e += gridDim.x) {
    const int en = e + gridDim.x;
    if (en < NE) {
      const float* znext = Z + (size_t)src[en] * KC + c;
#pragma unroll
      for (int k = 0; k < KG; ++k) __builtin_prefetch(znext + k * Cout, 0, 0);
    }
    const int s = src[e];
    const int d = dst[e];
    const float* zrow = Z + (size_t)s * KC + c;
    const float* wrow = W + (size_t)e * KG;
    float acc = 0.0f;
#pragma unroll
    for (int k = 0; k < KG; ++k) acc = fmaf(wrow[k], zrow[k * Cout], acc);
    atomicAdd(&agg[(size_t)d * Cout + c], acc);
  }
}

// --------------------------------------------------------------- finalizers
__global__ void finalize_relu(const float* __restrict__ agg, const float* __restrict__ R,
                              const float* __restrict__ bias, const float* __restrict__ deg,
                              float* __restrict__ out, int C) {
  size_t i = (size_t)blockIdx.x * blockDim.x + threadIdx.x;
  size_t n = (size_t)NN * C;
  if (i >= n) return;
  int node = (int)(i / C);
  int c = (int)(i - (size_t)node * C);
  float dv = fmaxf(deg[node], 1.0f);
  out[i] = fmaxf(agg[i] / dv + R[i] + bias[c], 0.0f);
}

__global__ void final_combine(const float* __restrict__ agg2, const float* __restrict__ R2,
                              const float* __restrict__ b2,
                              const float* __restrict__ aggs, const float* __restrict__ Rs,
                              const float* __restrict__ bs,
                              const float* __restrict__ deg, float* __restrict__ out) {
  size_t i = (size_t)blockIdx.x * blockDim.x + threadIdx.x;
  size_t n = (size_t)NN * 128;
  if (i >= n) return;
  int node = (int)(i >> 7);
  int c = (int)(i & 127);
  float dv = fmaxf(deg[node], 1.0f);
  float y = agg2[i] / dv + R2[i] + b2[c];
  float xs = aggs[i] / dv + Rs[i] + bs[c];
  out[i] = fmaxf(y + xs, 0.0f);
}

// ------------------------------------------------------------------- launch
template <int NC>
static inline void launch_gemm(const float* A, const float* B, float* C,
                               int M, hipStream_t s) {
  int tiles = ((M + 31) / 32) * (NC / 16);
  int blocks = (tiles + 7) / 8;  // 8 wave32 per 256-thread block
  gemm_k64_wmma_f32<NC><<<blocks, 256, 0, s>>>(A, B, C, M);
}

extern "C" void kernel_launch(void* const* d_in, const int* in_sizes, int n_in,
                              void* d_out, int out_size, void* d_ws, size_t ws_size,
                              hipStream_t stream) {
  const float* x      = (const float*)d_in[0];
  const int*   ei     = (const int*)  d_in[1];
  const float* pseudo = (const float*)d_in[2];
  const float* g1     = (const float*)d_in[3];
  const float* mu1    = (const float*)d_in[4];
  const float* sg1    = (const float*)d_in[5];
  const float* root1  = (const float*)d_in[6];
  const float* b1     = (const float*)d_in[7];
  const float* g2     = (const float*)d_in[8];
  const float* mu2    = (const float*)d_in[9];
  const float* sg2    = (const float*)d_in[10];
  const float* root2  = (const float*)d_in[11];
  const float* b2     = (const float*)d_in[12];
  const float* gs     = (const float*)d_in[13];
  const float* mus    = (const float*)d_in[14];
  const float* sgs    = (const float*)d_in[15];
  const float* roots  = (const float*)d_in[16];
  const float* bs     = (const float*)d_in[17];
  float* out = (float*)d_out;
  const int* src = ei;
  const int* dst = ei + NE;

  const size_t N = NN, E = NE;
  float* w = (float*)d_ws;
  size_t o = 0;
  float* deg  = w + o; o += N;
  float* agg1 = w + o; o += N * 64;
  float* agg2 = w + o; o += N * 128;
  float* aggs = w + o; o += N * 128;
  const size_t zeroCount = o;         // deg + all agg buffers, contiguous
  float* w1   = w + o; o += E * KG;
  float* w2   = w + o; o += E * KG;
  float* wsw  = w + o; o += E * KG;
  float* Z    = w + o; o += N * 1280; // reused for Z1 (640), Z2, Zs (1280)
  float* R1   = w + o; o += N * 64;
  float* H    = w + o; o += N * 64;
  float* R2   = w + o; o += N * 128;
  float* Rs   = w + o; o += N * 128;

  const int EDGE_GRID = 20480;

  // stage 0: init + edge weights + degree
  zero_f32<<<2048, 256, 0, stream>>>(deg, zeroCount);
  edge_weights_deg<<<(int)((E + 255) / 256), 256, 0, stream>>>(
      pseudo, dst, mu1, sg1, mu2, sg2, mus, sgs, w1, w2, wsw, deg);

  // conv1: Cin=64 -> Ch=64
  launch_gemm<640>(x, g1, Z, NN, stream);
  launch_gemm<64>(x, root1, R1, NN, stream);
  edge_aggregate<64><<<EDGE_GRID, 64, 0, stream>>>(Z, w1, src, dst, agg1);
  finalize_relu<<<(int)((N * 64 + 255) / 256), 256, 0, stream>>>(agg1, R1, b1, deg, H, 64);

  // conv2: Ch=64 -> Cout=128
  launch_gemm<1280>(H, g2, Z, NN, stream);
  launch_gemm<128>(H, root2, R2, NN, stream);
  edge_aggregate<128><<<EDGE_GRID, 128, 0, stream>>>(Z, w2, src, dst, agg2);

  // shortcut: Cin=64 -> Cout=128 (Z buffer reused; stream order protects it)
  launch_gemm<1280>(x, gs, Z, NN, stream);
  launch_gemm<128>(x, roots, Rs, NN, stream);
  edge_aggregate<128><<<EDGE_GRID, 128, 0, stream>>>(Z, wsw, src, dst, aggs);

  // out = relu(y + x_s)
  final_combine<<<(int)((N * 128 + 255) / 256), 256, 0, stream>>>(
      agg2, R2, b2, aggs, Rs, bs, deg, out);
}